// GCNDecoder_21388937134518
// MI455X (gfx1250) — compile-verified
//
#include <hip/hip_runtime.h>

#define N_NODES 50000
#define NE_EDGES 800000
#define BPAIRS 4096
#define DF 128
#define EPS_BN 1e-5f

typedef float v2f __attribute__((ext_vector_type(2)));
typedef float v8f __attribute__((ext_vector_type(8)));

__device__ __forceinline__ v8f wmma4(v2f a, v2f b, v8f c) {
  // V_WMMA_F32_16X16X4_F32 : D = A(16x4) x B(4x16) + C(16x16), full fp32
  return __builtin_amdgcn_wmma_f32_16x16x4_f32(
      /*neg_a=*/false, a, /*neg_b=*/false, b,
      /*c_mod=*/(short)0, c, /*reuse_a=*/false, /*reuse_b=*/false);
}

// ---------------- big GEMM: one wave computes a 16 x 128 strip ----------------
// C(Mx128) = A(MxK) @ Bt^T  (+bias), Bt is (128 x K) row-major (pre-transposed B).
// 8 independent v8f accumulators -> no WMMA->WMMA RAW hazards; A frag loaded once.
__global__ __launch_bounds__(256)
void wmma_gemm_strip(const float* __restrict__ A, const float* __restrict__ Bt,
                     const float* __restrict__ bias, float* __restrict__ C,
                     int M, int K) {
  int gtid = blockIdx.x * blockDim.x + threadIdx.x;
  int wave = gtid >> 5;                     // one wave per 16-row strip
  int lane = threadIdx.x & 31;
  if (wave * 16 >= M) return;               // wave-uniform: EXEC stays all-ones

  int l16   = lane & 15;
  int khalf = (lane >> 4) << 1;             // lanes 0-15 -> K {0,1}; 16-31 -> {2,3}
  const float* arow = A + (size_t)(wave * 16 + l16) * K + khalf;

  const float* brow[8];
#pragma unroll
  for (int nt = 0; nt < 8; ++nt)
    brow[nt] = Bt + (size_t)(nt * 16 + l16) * K + khalf;

  v8f zero = {};
  v8f acc[8];
#pragma unroll
  for (int nt = 0; nt < 8; ++nt) acc[nt] = zero;

  for (int k = 0; k < K; k += 4) {
    v2f a = *(const v2f*)(arow + k);
#pragma unroll
    for (int nt = 0; nt < 8; ++nt) {
      v2f b = *(const v2f*)(brow[nt] + k);
      acc[nt] = wmma4(a, b, acc[nt]);
    }
  }

  int rbase = wave * 16 + ((lane >> 4) << 3);     // M = r + 8*(lane>=16)
#pragma unroll
  for (int nt = 0; nt < 8; ++nt) {
    float bs = bias ? bias[nt * 16 + l16] : 0.f;
    float* crow = C + (size_t)rbase * 128 + nt * 16 + l16;
#pragma unroll
    for (int r = 0; r < 8; ++r)
      crow[(size_t)r * 128] = acc[nt][r] + bs;
  }
}

// ---------------- small GEMM: one wave per 16x16 tile (NN / NT) ----------------
__global__ __launch_bounds__(256)
void wmma_matmul(const float* __restrict__ A, const float* __restrict__ Bm,
                 const float* __restrict__ bias, float* __restrict__ C,
                 int M, int Nc, int K, int bTrans) {
  int gtid = blockIdx.x * blockDim.x + threadIdx.x;
  int wave = gtid >> 5;
  int lane = threadIdx.x & 31;
  int ntiles = Nc >> 4;
  int mt = wave / ntiles;
  int nt = wave - mt * ntiles;
  if (mt * 16 >= M) return;

  int l16   = lane & 15;
  int khalf = (lane >> 4) << 1;
  const float* arow = A + (size_t)(mt * 16 + l16) * K + khalf;

  v8f acc = {};
  if (bTrans) {
    const float* brow = Bm + (size_t)(nt * 16 + l16) * K + khalf;
    for (int k = 0; k < K; k += 4) {
      v2f a = *(const v2f*)(arow + k);
      v2f b = *(const v2f*)(brow + k);
      acc = wmma4(a, b, acc);
    }
  } else {
    const float* bcol = Bm + (size_t)khalf * Nc + nt * 16 + l16;
    for (int k = 0; k < K; k += 4) {
      v2f a = *(const v2f*)(arow + k);
      v2f b;
      b.x = bcol[(size_t)k * Nc];
      b.y = bcol[(size_t)(k + 1) * Nc];
      acc = wmma4(a, b, acc);
    }
  }

  float bs = bias ? bias[nt * 16 + l16] : 0.f;
  int rbase = mt * 16 + ((lane >> 4) << 3);
  float* crow = C + (size_t)rbase * Nc + nt * 16 + l16;
#pragma unroll
  for (int r = 0; r < 8; ++r)
    crow[(size_t)r * Nc] = acc[r] + bs;
}

// ---------------- CSR build (once per call; reused by all 3 layers) ------------
__global__ void zero_int(int* p, int n) {
  int t = blockIdx.x * blockDim.x + threadIdx.x;
  if (t < n) p[t] = 0;
}

__global__ void count_edges(const int* __restrict__ ecol, int* __restrict__ cnt) {
  int e = blockIdx.x * blockDim.x + threadIdx.x;
  if (e < NE_EDGES) atomicAdd(&cnt[ecol[e]], 1);
}

// single-block exclusive scan of cnt[N] -> rowptr[N+1]
__global__ __launch_bounds__(1024)
void scan_rowptr(const int* __restrict__ cnt, int* __restrict__ rowptr) {
  __shared__ int part[1024];
  const int CH = (N_NODES + 1023) / 1024;   // 49 elements per thread
  int tid = threadIdx.x;
  int base = tid * CH;
  int s = 0;
  for (int i = 0; i < CH; ++i) {
    int idx = base + i;
    if (idx < N_NODES) s += cnt[idx];
  }
  part[tid] = s;
  __syncthreads();
  for (int off = 1; off < 1024; off <<= 1) {   // Hillis-Steele inclusive scan
    int v = 0;
    if (tid >= off) v = part[tid - off];
    __syncthreads();
    if (tid >= off) part[tid] += v;
    __syncthreads();
  }
  int run = (tid == 0) ? 0 : part[tid - 1];    // exclusive prefix for this chunk
  for (int i = 0; i < CH; ++i) {
    int idx = base + i;
    if (idx < N_NODES) { rowptr[idx] = run; run += cnt[idx]; }
  }
  if (tid == 1023) rowptr[N_NODES] = run;      // total edge count
}

__global__ void copy_int(const int* __restrict__ src, int* __restrict__ dst, int n) {
  int t = blockIdx.x * blockDim.x + threadIdx.x;
  if (t < n) dst[t] = src[t];
}

__global__ void csr_fill(const int* __restrict__ erow, const int* __restrict__ ecol,
                         int* __restrict__ cursor, int* __restrict__ srcidx) {
  int e = blockIdx.x * blockDim.x + threadIdx.x;
  if (e >= NE_EDGES) return;
  int c = ecol[e];
  int p = atomicAdd(&cursor[c], 1);
  srcidx[p] = erow[e];
}

__global__ void dis_from_cnt(const int* __restrict__ cnt, float* __restrict__ dis, int n) {
  int t = blockIdx.x * blockDim.x + threadIdx.x;
  if (t < n) dis[t] = rsqrtf((float)cnt[t] + 1.0f);   // +1 = self-loop
}

// ---------------- atomic-free aggregation: one wave per destination node ------
__global__ __launch_bounds__(256)
void gcn_aggregate(const int* __restrict__ rowptr, const int* __restrict__ srcidx,
                   const float* __restrict__ dis, const float* __restrict__ lin,
                   float* __restrict__ agg) {
  int t = blockIdx.x * blockDim.x + threadIdx.x;
  int c = t >> 5;
  if (c >= N_NODES) return;
  int lane = threadIdx.x & 31;
  float dc = dis[c];
  const float4 sv = *(const float4*)(lin + (size_t)c * DF + lane * 4);
  float s = dc * dc;                       // self-loop norm
  float ax = s * sv.x, ay = s * sv.y, az = s * sv.z, aw = s * sv.w;
  int e0 = rowptr[c], e1 = rowptr[c + 1];
  for (int j = e0; j < e1; ++j) {
    int r = srcidx[j];                     // broadcast scalar load
    float nrm = dis[r] * dc;
    const float4 v = *(const float4*)(lin + (size_t)r * DF + lane * 4);
    ax += nrm * v.x; ay += nrm * v.y; az += nrm * v.z; aw += nrm * v.w;
  }
  float4 o; o.x = ax; o.y = ay; o.z = az; o.w = aw;
  *(float4*)(agg + (size_t)c * DF + lane * 4) = o;
}

// ---------------- batch norm + leaky ----------------
__global__ void zero256(float* p) { p[threadIdx.x] = 0.0f; }

__global__ __launch_bounds__(128)
void bn_stats(const float* __restrict__ agg, float* __restrict__ stats, int n) {
  int f = threadIdx.x;
  int r0 = blockIdx.x * 512;
  int r1 = r0 + 512; if (r1 > n) r1 = n;
  float s = 0.f, s2 = 0.f;
  for (int r = r0; r < r1; ++r) {
    float v = agg[(size_t)r * DF + f];
    s += v; s2 += v * v;
  }
  atomicAdd(&stats[f], s);
  atomicAdd(&stats[DF + f], s2);
}

__global__ void bn_finalize(float* __restrict__ stats, const float* __restrict__ g,
                            const float* __restrict__ be, float inv_n) {
  int f = threadIdx.x;                      // 128 threads
  float m   = stats[f] * inv_n;
  float var = stats[DF + f] * inv_n - m * m;
  float sc  = g[f] * rsqrtf(var + EPS_BN);
  stats[f]      = sc;
  stats[DF + f] = be[f] - m * sc;
}

__global__ __launch_bounds__(256)
void bn_apply_leaky(const float* __restrict__ agg, const float* __restrict__ stats,
                    float* __restrict__ h) {
  int t = blockIdx.x * blockDim.x + threadIdx.x;
  if (t >= N_NODES * DF) return;
  int f = t & (DF - 1);
  float y = agg[t] * stats[f] + stats[DF + f];
  h[t] = (y >= 0.f) ? y : 0.1f * y;
}

// ---------------- misc ----------------
__global__ void transpose128(const float* __restrict__ src, float* __restrict__ dst) {
  int t = blockIdx.x * blockDim.x + threadIdx.x;   // 16384 threads
  if (t >= 128 * 128) return;
  int r = t >> 7, c = t & 127;
  dst[c * 128 + r] = src[t];
}

__global__ __launch_bounds__(256)
void gather_pairs(const float* __restrict__ h, const int* __restrict__ di,
                  float* __restrict__ amat, float* __restrict__ bmat) {
  int t = blockIdx.x * blockDim.x + threadIdx.x;
  if (t >= BPAIRS * 32) return;
  int j = t >> 5, q = t & 31;
  int ia = di[2 * j] - 1;     if (ia < 0) ia += N_NODES;   // JAX wraparound
  int ib = di[2 * j + 1] - 1; if (ib < 0) ib += N_NODES;
  *(float4*)(amat + (size_t)j * DF + q * 4) = *(const float4*)(h + (size_t)ia * DF + q * 4);
  *(float4*)(bmat + (size_t)j * DF + q * 4) = *(const float4*)(h + (size_t)ib * DF + q * 4);
}

__global__ __launch_bounds__(256)
void rowdot(const float* __restrict__ tm, const float* __restrict__ bm,
            float* __restrict__ out) {
  int t = blockIdx.x * blockDim.x + threadIdx.x;
  int j = t >> 5, lane = t & 31;
  if (j >= BPAIRS) return;
  float4 tv = *(const float4*)(tm + (size_t)j * DF + lane * 4);
  float4 bv = *(const float4*)(bm + (size_t)j * DF + lane * 4);
  float p = tv.x * bv.x + tv.y * bv.y + tv.z * bv.z + tv.w * bv.w;
#pragma unroll
  for (int off = 16; off > 0; off >>= 1)
    p += __shfl_down(p, off, 32);
  if (lane == 0) out[j] = p;
}

extern "C" void kernel_launch(void* const* d_in, const int* in_sizes, int n_in,
                              void* d_out, int out_size, void* d_ws, size_t ws_size,
                              hipStream_t stream) {
  const float* x   = (const float*)d_in[0];
  const int*   edg = (const int*)d_in[1];
  const int*   di  = (const int*)d_in[2];
  const float* w1  = (const float*)d_in[3];
  const float* b1  = (const float*)d_in[4];
  const float* w2  = (const float*)d_in[5];
  const float* b2  = (const float*)d_in[6];
  const float* w3  = (const float*)d_in[7];
  const float* b3  = (const float*)d_in[8];
  const float* g1  = (const float*)d_in[9];
  const float* be1 = (const float*)d_in[10];
  const float* g2  = (const float*)d_in[11];
  const float* be2 = (const float*)d_in[12];
  const float* g3  = (const float*)d_in[13];
  const float* be3 = (const float*)d_in[14];
  const float* p1  = (const float*)d_in[15];
  const float* p2  = (const float*)d_in[16];
  float* out = (float*)d_out;

  float* ws = (float*)d_ws;
  size_t off = 0;
  float* dis    = ws + off; off += N_NODES;        // 50000
  float* stats  = ws + off; off += 256;
  float* Wt     = ws + off; off += 128 * 128;      // transposed weight (reused)
  int*   cnt    = (int*)(ws + off); off += N_NODES;
  int*   rowptr = (int*)(ws + off); off += N_NODES + 16;
  int*   cursor = (int*)(ws + off); off += N_NODES;
  int*   srcidx = (int*)(ws + off); off += NE_EDGES;
  float* lin    = ws + off; off += (size_t)N_NODES * DF;
  float* agg    = ws + off; off += (size_t)N_NODES * DF;
  float* h      = ws + off; off += (size_t)N_NODES * DF;
  // decoder scratch overlays lin/agg (free after the GNN layers):
  float* U    = lin;                               // 128*128
  float* Qt   = lin + 16384;                       // 128*128 (= Q^T)
  float* amat = lin + 32768;                       // B*DF
  float* bmat = amat + (size_t)BPAIRS * DF;        // B*DF
  float* tmat = bmat + (size_t)BPAIRS * DF;        // B*DF

  const int* erow = edg;
  const int* ecol = edg + NE_EDGES;

  // ---- build CSR of incoming edges once; degrees fall out for free ----
  zero_int<<<(N_NODES + 255) / 256, 256, 0, stream>>>(cnt, N_NODES);
  count_edges<<<(NE_EDGES + 255) / 256, 256, 0, stream>>>(ecol, cnt);
  scan_rowptr<<<1, 1024, 0, stream>>>(cnt, rowptr);
  copy_int<<<(N_NODES + 255) / 256, 256, 0, stream>>>(rowptr, cursor, N_NODES);
  csr_fill<<<(NE_EDGES + 255) / 256, 256, 0, stream>>>(erow, ecol, cursor, srcidx);
  dis_from_cnt<<<(N_NODES + 255) / 256, 256, 0, stream>>>(cnt, dis, N_NODES);

  const float* Wl[3] = {w1, w2, w3};
  const float* Bl[3] = {b1, b2, b3};
  const float* Gl[3] = {g1, g2, g3};
  const float* El[3] = {be1, be2, be3};
  const float* in = x;
  const int strip_waves = N_NODES / 16;            // 3125 waves
  for (int l = 0; l < 3; ++l) {
    transpose128<<<64, 256, 0, stream>>>(Wl[l], Wt);
    wmma_gemm_strip<<<(strip_waves * 32 + 255) / 256, 256, 0, stream>>>(
        in, Wt, Bl[l], lin, N_NODES, DF);
    gcn_aggregate<<<(N_NODES * 32 + 255) / 256, 256, 0, stream>>>(
        rowptr, srcidx, dis, lin, agg);
    zero256<<<1, 256, 0, stream>>>(stats);
    bn_stats<<<(N_NODES + 511) / 512, 128, 0, stream>>>(agg, stats, N_NODES);
    bn_finalize<<<1, 128, 0, stream>>>(stats, Gl[l], El[l], 1.0f / (float)N_NODES);
    bn_apply_leaky<<<(N_NODES * DF + 255) / 256, 256, 0, stream>>>(agg, stats, h);
    in = h;
  }

  // ---- decoder: Qt = Q^T = p1 @ p2^T @ p1^T (two NT GEMMs, no transpose pass)
  const int small_tiles = (DF / 16) * (DF / 16);   // 64 waves
  wmma_matmul<<<(small_tiles * 32 + 255) / 256, 256, 0, stream>>>(p1, p2, nullptr, U, DF, DF, DF, 1);
  wmma_matmul<<<(small_tiles * 32 + 255) / 256, 256, 0, stream>>>(U, p1, nullptr, Qt, DF, DF, DF, 1);
  gather_pairs<<<(BPAIRS * 32 + 255) / 256, 256, 0, stream>>>(h, di, amat, bmat);
  wmma_gemm_strip<<<((BPAIRS / 16) * 32 + 255) / 256, 256, 0, stream>>>(
      amat, Qt, nullptr, tmat, BPAIRS, DF);
  rowdot<<<(BPAIRS * 32 + 255) / 256, 256, 0, stream>>>(tmat, bmat, out);
}